// Classifier_61040075211449
// MI455X (gfx1250) — compile-verified
//
#include <hip/hip_runtime.h>
#include <hip/hip_bf16.h>
#include <stdint.h>

// ---------------------------------------------------------------------------
// GNN SimpleConv(mean, self_loop) -> threshold col 0.
// Mathematically reduced: out[i] = (x[i,0] + sum_{e:dst=i} x[src[e],0]) > 0
// (mean>0 <=> sum>0 since deg>=1; only feature column 0 survives).
// Memory-bound scatter-add. CDNA5 paths: GLOBAL_LOAD_ASYNC_TO_LDS_B32
// double-buffered edge-index staging fenced by s_wait_asynccnt (ASYNCcnt),
// native no-return global_atomic_add_f32 for the segment sum.
// ---------------------------------------------------------------------------

#ifndef __has_builtin
#define __has_builtin(x) 0
#endif

#if __has_builtin(__builtin_amdgcn_global_load_async_to_lds_b32)
#define USE_ASYNC_LDS 1
#else
#define USE_ASYNC_LDS 0
#endif

// Builtin pointer params are LangAS-qualified: global = addrspace(1),
// LDS = addrspace(3) (confirmed by hipcc diagnostics in earlier round).
typedef __attribute__((address_space(1))) int*   as1_i32p;
typedef __attribute__((address_space(3))) int*   as3_i32p;
typedef __attribute__((address_space(1))) float* as1_f32p;

#define TPB   256          // 8 wave32 per block
#define TILE  1024         // edges staged per buffer
#define EPT   (TILE / TPB) // 4 elements per thread -> 4 async instr per array
#define D_FEAT 64

__device__ __forceinline__ void async_idx_load(int* lds_dst, const int* gsrc) {
#if USE_ASYNC_LDS
  // gfx1250: global_load_async_to_lds_b32 (per-lane 4B, tracked by ASYNCcnt)
  __builtin_amdgcn_global_load_async_to_lds_b32((as1_i32p)(int*)gsrc,
                                                (as3_i32p)lds_dst, 0, 0);
#else
  *lds_dst = *gsrc;
#endif
}

template <int N>
__device__ __forceinline__ void wait_async() {
#if USE_ASYNC_LDS
#if __has_builtin(__builtin_amdgcn_s_wait_asynccnt)
  __builtin_amdgcn_s_wait_asynccnt(N);
#else
  asm volatile("s_wait_asynccnt %0" ::"i"(N) : "memory");
#endif
#endif
}

__device__ __forceinline__ void atomic_add_f32(float* addr, float v) {
#if __has_builtin(__builtin_amdgcn_global_atomic_fadd_f32)
  // Result discarded -> backend emits no-return global_atomic_add_f32
  // (STOREcnt path, resolved at the L2 atomic units).
  (void)__builtin_amdgcn_global_atomic_fadd_f32((as1_f32p)addr, v);
#else
  (void)__hip_atomic_fetch_add(addr, v, __ATOMIC_RELAXED,
                               __HIP_MEMORY_SCOPE_AGENT);
#endif
}

__global__ void __launch_bounds__(TPB)
edge_scatter(const int* __restrict__ src, const int* __restrict__ dst,
             const float* __restrict__ x, float* __restrict__ acc,
             int ne, int ntiles) {
  __shared__ int s_src[2][TILE];
  __shared__ int s_dst[2][TILE];

  int t = blockIdx.x;
  if (t >= ntiles) return;

  // Issue one tile's worth of async index loads (clamped so every wave
  // issues exactly 2*EPT async instructions -> deterministic ASYNCcnt).
  auto issue = [&](int buf, int tile) {
    const int base = tile * TILE;
#pragma unroll
    for (int k = 0; k < EPT; ++k) {
      const int i = threadIdx.x + k * TPB;
      int e = base + i;
      if (e >= ne) e = ne - 1;  // clamp: redundant load, uniform issue count
      async_idx_load(&s_src[buf][i], src + e);
      async_idx_load(&s_dst[buf][i], dst + e);
    }
  };

  issue(0, t);                 // prologue: ASYNCcnt = 2*EPT
  int buf = 0;
  while (t < ntiles) {
    const int nt = t + gridDim.x;
    if (nt < ntiles) {
      issue(buf ^ 1, nt);      // ASYNCcnt = 4*EPT
      wait_async<2 * EPT>();   // oldest 2*EPT (current tile) retired, in order
    } else {
      wait_async<0>();
    }
    __syncthreads();           // every wave's stage of this tile is complete

    const int base = t * TILE;
#pragma unroll
    for (int k = 0; k < EPT; ++k) {
      const int i = threadIdx.x + k * TPB;
      if (base + i < ne) {
        const int s = s_src[buf][i];
        const int d = s_dst[buf][i];
        const float v = x[(size_t)s * D_FEAT];  // random gather, L2-resident
        atomic_add_f32(&acc[d], v);
      }
    }
    __syncthreads();           // done reading buf before it is re-staged
    buf ^= 1;
    t = nt;
  }
}

__global__ void __launch_bounds__(TPB)
finish(const float* __restrict__ x, const float* __restrict__ acc,
       float* __restrict__ out, int n) {
  const int i = blockIdx.x * TPB + threadIdx.x;
  if (i < n) {
    const float s = acc[i] + x[(size_t)i * D_FEAT];  // fold in self-loop
    out[i] = (s > 0.0f) ? 1.0f : 0.0f;
  }
}

extern "C" void kernel_launch(void* const* d_in, const int* in_sizes, int n_in,
                              void* d_out, int out_size, void* d_ws, size_t ws_size,
                              hipStream_t stream) {
  const float* x  = (const float*)d_in[0];
  const int*   ei = (const int*)d_in[1];   // harness: integer inputs as int32

  const int n_nodes = in_sizes[0] / D_FEAT;
  const int n_edges = in_sizes[1] / 2;
  const int* src = ei;                      // edge_index[0]
  const int* dst = ei + n_edges;            // edge_index[1]

  float* acc = (float*)d_ws;                // n_nodes floats of scratch
  (void)hipMemsetAsync(acc, 0, (size_t)n_nodes * sizeof(float), stream);

  const int ntiles = (n_edges + TILE - 1) / TILE;
  const int grid   = ntiles < 1024 ? ntiles : 1024;  // persistent, grid-stride over tiles
  edge_scatter<<<grid, TPB, 0, stream>>>(src, dst, x, acc, n_edges, ntiles);

  const int fgrid = (n_nodes + TPB - 1) / TPB;
  finish<<<fgrid, TPB, 0, stream>>>(x, acc, (float*)d_out, n_nodes);
}